// PointSAModuleMsg_20950850469944
// MI455X (gfx1250) — compile-verified
//
#include <hip/hip_runtime.h>
#include <float.h>

// ---------------- problem constants (from reference) ----------------
#define N_PTS   16384
#define M_CTR   4096          // N * 0.25
#define K_NB    16            // NSAMPLE
#define KD      32            // max K*dil (dil=2)
#define F_INC   64            // input feature channels
#define C0      67            // F_IN + 3
#define C0P     68            // padded K for GEMM (multiple of 4)
#define C1      64
#define C2      128
#define NEDGE   (M_CTR * K_NB)   // 65536 rows in the edge-feature matrix
#define BN_EPS  1e-5f

typedef float v2f __attribute__((ext_vector_type(2)));
typedef float v8f __attribute__((ext_vector_type(8)));

// =====================================================================
// 1) Farthest point sampling: one persistent workgroup, whole cloud in
//    LDS (3*64KB pos + 64KB mind = 256KB of the WGP's 320KB).
// =====================================================================
__global__ __launch_bounds__(1024) void fps_kernel(const float* __restrict__ pos,
                                                   int* __restrict__ idx) {
  extern __shared__ float smem[];
  float* px   = smem;
  float* py   = px + N_PTS;
  float* pz   = py + N_PTS;
  float* mind = pz + N_PTS;
  float* redV = mind + N_PTS;          // 32 per-wave maxima
  int*   redI = (int*)(redV + 32);     // 32 per-wave arg indices
  int*   curS = redI + 32;             // broadcast slot

  const int tid  = threadIdx.x;
  const int lane = tid & 31;
  const int wave = tid >> 5;

  for (int t = tid; t < N_PTS; t += 1024) {
    px[t] = pos[3 * t + 0];
    py[t] = pos[3 * t + 1];
    pz[t] = pos[3 * t + 2];
    mind[t] = 1e30f;
  }
  if (tid == 0) { idx[0] = 0; *curS = 0; }
  __syncthreads();

  int cur = 0;
  for (int s = 1; s < M_CTR; ++s) {
    const float cx = px[cur], cy = py[cur], cz = pz[cur];
    float bv = -1.0f;
    int   bi = N_PTS;
    #pragma unroll
    for (int u = 0; u < N_PTS / 1024; ++u) {    // 16 points / thread
      const int t = tid + u * 1024;
      float dx = px[t] - cx, dy = py[t] - cy, dz = pz[t] - cz;
      float d  = dx * dx + dy * dy + dz * dz;
      float nm = fminf(mind[t], d);
      mind[t] = nm;
      if (nm > bv) { bv = nm; bi = t; }         // strict > keeps lowest idx
    }
    // wave32 argmax reduction (value desc, index asc on ties)
    for (int off = 16; off > 0; off >>= 1) {
      float ov = __shfl_down(bv, off, 32);
      int   oi = __shfl_down(bi, off, 32);
      if (ov > bv || (ov == bv && oi < bi)) { bv = ov; bi = oi; }
    }
    if (lane == 0) { redV[wave] = bv; redI[wave] = bi; }
    __syncthreads();
    if (wave == 0) {
      float v  = redV[lane];
      int   i2 = redI[lane];
      for (int off = 16; off > 0; off >>= 1) {
        float ov = __shfl_down(v, off, 32);
        int   oi = __shfl_down(i2, off, 32);
        if (ov > v || (ov == v && oi < i2)) { v = ov; i2 = oi; }
      }
      if (lane == 0) { *curS = i2; idx[s] = i2; }
    }
    __syncthreads();
    cur = *curS;
  }
}

// =====================================================================
// 2) Gather sampled centers; also emit pos_s and batch tails of d_out.
// =====================================================================
__global__ void gather_kernel(const float* __restrict__ pos,
                              const int* __restrict__ idx,
                              float* __restrict__ pos_s,
                              float* __restrict__ out) {
  int m = blockIdx.x * blockDim.x + threadIdx.x;
  if (m >= M_CTR) return;
  int j = idx[m];
  float x = pos[3 * j + 0], y = pos[3 * j + 1], z = pos[3 * j + 2];
  pos_s[3 * m + 0] = x; pos_s[3 * m + 1] = y; pos_s[3 * m + 2] = z;
  float* tail = out + (long)M_CTR * (2 * C2);
  tail[3 * m + 0] = x; tail[3 * m + 1] = y; tail[3 * m + 2] = z;
  tail[3 * M_CTR + m] = 0.0f;   // batch is all-zero int32; bit pattern == 0.0f
}

// =====================================================================
// 3) KNN: one block per center, d^2 (64KB) in LDS, 32 argmin passes.
// =====================================================================
__global__ __launch_bounds__(256) void knn_kernel(const float* __restrict__ pos,
                                                  const float* __restrict__ pos_s,
                                                  int* __restrict__ nb) {
  extern __shared__ float smem[];
  float* d2   = smem;               // N_PTS
  float* redV = d2 + N_PTS;         // 8
  int*   redI = (int*)(redV + 8);   // 8

  const int m   = blockIdx.x;
  const int tid = threadIdx.x;
  const int lane = tid & 31;
  const int wave = tid >> 5;
  const float cx = pos_s[3 * m + 0], cy = pos_s[3 * m + 1], cz = pos_s[3 * m + 2];

  for (int t = tid; t < N_PTS; t += 256) {
    float dx = pos[3 * t + 0] - cx;
    float dy = pos[3 * t + 1] - cy;
    float dz = pos[3 * t + 2] - cz;
    d2[t] = dx * dx + dy * dy + dz * dz;
  }
  __syncthreads();

  for (int kk = 0; kk < KD; ++kk) {
    float bv = FLT_MAX;
    int   bi = N_PTS;
    for (int t = tid; t < N_PTS; t += 256) {    // 64 candidates / thread
      float v = d2[t];
      if (v < bv) { bv = v; bi = t; }
    }
    for (int off = 16; off > 0; off >>= 1) {
      float ov = __shfl_down(bv, off, 32);
      int   oi = __shfl_down(bi, off, 32);
      if (ov < bv || (ov == bv && oi < bi)) { bv = ov; bi = oi; }
    }
    if (lane == 0) { redV[wave] = bv; redI[wave] = bi; }
    __syncthreads();
    if (tid == 0) {
      float v  = redV[0];
      int   i2 = redI[0];
      for (int w = 1; w < 8; ++w)
        if (redV[w] < v || (redV[w] == v && redI[w] < i2)) { v = redV[w]; i2 = redI[w]; }
      nb[(long)m * KD + kk] = i2;
      d2[i2] = FLT_MAX;           // exclude from later passes
    }
    __syncthreads();
  }
}

// =====================================================================
// 4) Build edge features F[e][0:68]: [x_j | pos_j - pos_i | 0-pad]
// =====================================================================
__global__ __launch_bounds__(128) void feat_kernel(const float* __restrict__ x,
                                                   const float* __restrict__ pos,
                                                   const float* __restrict__ pos_s,
                                                   const int* __restrict__ nb,
                                                   float* __restrict__ F,
                                                   int dil) {
  const int e = blockIdx.x;        // 0..NEDGE-1
  const int m = e >> 4, k = e & 15;
  int slot;
  if (dil > 1) {                   // deterministic hash subsample of the 32-NN
    unsigned u = (unsigned)(m * 16 + k);
    u = u * 2654435761u + 42u; u ^= u >> 16; u *= 2246822519u; u ^= u >> 13;
    slot = (int)(u & 31u);
  } else {
    slot = k;                      // first 16 of the 32 nearest == 16-NN
  }
  const int j = nb[(long)m * KD + slot];
  const int c = threadIdx.x;
  if (c >= C0P) return;
  float v;
  if (c < F_INC)      v = x[(long)j * F_INC + c];
  else if (c < C0)    v = pos[3 * j + (c - F_INC)] - pos_s[3 * m + (c - F_INC)];
  else                v = 0.0f;    // K padding column
  F[(long)e * C0P + c] = v;
}

// =====================================================================
// 5) Zero-pad a Kdim x Ncols weight into C0P x Ncols so GEMM loads
//    need no bounds checks (keeps the WMMA K-loop branch-free).
// =====================================================================
__global__ void pad_b_kernel(const float* __restrict__ W, float* __restrict__ Bp,
                             int Kdim, int Ncols) {
  int i = blockIdx.x * blockDim.x + threadIdx.x;
  if (i >= C0P * Ncols) return;
  int r = i / Ncols, c = i - r * Ncols;
  Bp[i] = (r < Kdim) ? W[r * Ncols + c] : 0.0f;
}

// =====================================================================
// 6) WMMA fp32 GEMM: each wave computes one 16-row strip across the
//    FULL output width (NT 16-col tiles), so one A fragment feeds NT
//    v_wmma_f32_16x16x4_f32 ops per K-chunk. Branch-free inner loop
//    (weights pre-padded; FUSE is a compile-time flag that applies
//    relu(a*scale[k]+shift[k]) to A on load = fused BN+ReLU).
//    K-loop kept rolled (unroll 1): the clause of B loads + staged
//    s_wait_loadcnt already hides latency, and unrolling only forces
//    accumulator phi-copies (v_dual_mov storms) between iterations.
//    Also accumulates per-column sum/sumsq for BatchNorm stats.
// =====================================================================
template <int KCH, int NT, bool FUSE>
__global__ __launch_bounds__(256) void wmma_gemm_kernel(
    const float* __restrict__ A, int lda,
    const float* __restrict__ B,                 // [KCH*4, NT*16], padded
    const float* __restrict__ bias,
    const float* __restrict__ aScale, const float* __restrict__ aShift,
    float* __restrict__ C,                       // [NEDGE, NT*16]
    float* __restrict__ colSum, float* __restrict__ colSq) {
  constexpr int NCOLS = NT * 16;
  const int wave = threadIdx.x >> 5;
  const int lane = threadIdx.x & 31;
  const int r0 = (blockIdx.x * 8 + wave) * 16;   // grid sized exactly: no guard

  // ISA A layout (16x4 f32): lanes 0-15 -> row=lane, K={0,1}; lanes 16-31 -> K={2,3}
  const int arow = lane & 15;
  const int kh   = (lane >> 4) * 2;
  const int bcol = lane & 15;
  const float* Arow = A + (long)(r0 + arow) * lda;

  v8f acc[NT];
  #pragma unroll
  for (int t = 0; t < NT; ++t) acc[t] = (v8f){};

  #pragma unroll 1
  for (int kc = 0; kc < KCH; ++kc) {
    const int k0 = kc * 4 + kh;
    __builtin_prefetch(Arow + kc * 4 + 16, 0, 1);   // next K-chunks of A strip
    float a0 = Arow[k0];
    float a1 = Arow[k0 + 1];
    if (FUSE) {                                  // fused BN + ReLU of layer-1 output
      a0 = fmaxf(a0 * aScale[k0] + aShift[k0], 0.0f);
      a1 = fmaxf(a1 * aScale[k0 + 1] + aShift[k0 + 1], 0.0f);
    }
    v2f a; a.x = a0; a.y = a1;
    const float* Bk = B + (long)k0 * NCOLS + bcol;
    #pragma unroll
    for (int t = 0; t < NT; ++t) {
      v2f b;
      b.x = Bk[t * 16];
      b.y = Bk[NCOLS + t * 16];
      acc[t] = __builtin_amdgcn_wmma_f32_16x16x4_f32(false, a, false, b,
                                                     (short)0, acc[t], false, false);
    }
  }

  // C/D layout: lanes 0-15 rows 0..7, lanes 16-31 rows 8..15; col = lane&15
  const int rbase = (lane >> 4) * 8;
  #pragma unroll
  for (int t = 0; t < NT; ++t) {
    const int col = t * 16 + bcol;
    const float bn = bias[col];
    float s = 0.0f, sq = 0.0f;
    #pragma unroll
    for (int v = 0; v < 8; ++v) {
      float val = acc[t][v] + bn;
      s  += val;
      sq += val * val;
      C[(long)(r0 + rbase + v) * NCOLS + col] = val;
    }
    atomicAdd(&colSum[col], s);
    atomicAdd(&colSq[col], sq);
  }
}

// =====================================================================
// 7) Fold BN stats into per-column affine: y = x*sc + sh
// =====================================================================
__global__ void stats_kernel(const float* __restrict__ colSum,
                             const float* __restrict__ colSq,
                             const float* __restrict__ gamma,
                             const float* __restrict__ beta,
                             float* __restrict__ sc, float* __restrict__ sh,
                             int Ccols) {
  int c = blockIdx.x * blockDim.x + threadIdx.x;
  if (c >= Ccols) return;
  const float cnt  = (float)NEDGE;
  const float mean = colSum[c] / cnt;
  const float var  = colSq[c] / cnt - mean * mean;
  const float g    = gamma[c] * rsqrtf(var + BN_EPS);
  sc[c] = g;
  sh[c] = beta[c] - mean * g;
}

// =====================================================================
// 8) out = max_k relu(bn2(H2) + bnd(Hd)) per center, write x_new slice.
// =====================================================================
__global__ __launch_bounds__(128) void combine_kernel(
    const float* __restrict__ H2, const float* __restrict__ Hd,
    const float* __restrict__ s2, const float* __restrict__ t2,
    const float* __restrict__ sd, const float* __restrict__ td,
    float* __restrict__ out, int scaleOff) {
  const int m = blockIdx.x;
  const int n = threadIdx.x;
  const float a = s2[n], b = t2[n], c = sd[n], d = td[n];
  float mx = -FLT_MAX;
  #pragma unroll
  for (int k = 0; k < K_NB; ++k) {
    const long e = (long)(m * K_NB + k);
    float v = fmaxf(H2[e * C2 + n] * a + b + Hd[e * C2 + n] * c + d, 0.0f);
    mx = fmaxf(mx, v);
  }
  out[(long)m * (2 * C2) + scaleOff + n] = mx;
}

// =====================================================================
// host-side orchestration
// =====================================================================
extern "C" void kernel_launch(void* const* d_in, const int* in_sizes, int n_in,
                              void* d_out, int out_size, void* d_ws, size_t ws_size,
                              hipStream_t stream) {
  const float* x   = (const float*)d_in[0];
  const float* pos = (const float*)d_in[1];
  // d_in[2] = batch (all zeros, int32) — not needed
  const float* w1  = (const float*)d_in[3];
  const float* b1  = (const float*)d_in[4];
  const float* g1  = (const float*)d_in[5];
  const float* bt1 = (const float*)d_in[6];
  const float* w2  = (const float*)d_in[7];
  const float* b2  = (const float*)d_in[8];
  const float* g2  = (const float*)d_in[9];
  const float* bt2 = (const float*)d_in[10];
  const float* wd  = (const float*)d_in[11];
  const float* bd  = (const float*)d_in[12];
  const float* gd  = (const float*)d_in[13];
  const float* btd = (const float*)d_in[14];
  float* out = (float*)d_out;

  // ---- workspace carve-up (~102 MB) ----
  char* ws = (char*)d_ws;
  size_t off = 0;
  auto take = [&](size_t bytes) {
    char* p = ws + off;
    off = (off + bytes + 255) & ~(size_t)255;
    return p;
  };
  int*   idx   = (int*)  take((size_t)M_CTR * sizeof(int));
  int*   nb32  = (int*)  take((size_t)M_CTR * KD * sizeof(int));
  float* pos_s = (float*)take((size_t)M_CTR * 3 * sizeof(float));
  float* F     = (float*)take((size_t)NEDGE * C0P * sizeof(float));
  float* H1    = (float*)take((size_t)NEDGE * C1 * sizeof(float));
  float* H2    = (float*)take((size_t)NEDGE * C2 * sizeof(float));
  float* Hd    = (float*)take((size_t)NEDGE * C2 * sizeof(float));
  float* Bp1   = (float*)take((size_t)C0P * C1 * sizeof(float));
  float* Bpd   = (float*)take((size_t)C0P * C2 * sizeof(float));
  float* stats = (float*)take((size_t)12 * C2 * sizeof(float));
  float* sum1 = stats + 0 * C2,  *sq1 = stats + 1 * C2;
  float* sum2 = stats + 2 * C2,  *sq2 = stats + 3 * C2;
  float* sumd = stats + 4 * C2,  *sqd = stats + 5 * C2;
  float* s1v  = stats + 6 * C2,  *t1v = stats + 7 * C2;
  float* s2v  = stats + 8 * C2,  *t2v = stats + 9 * C2;
  float* sdv  = stats + 10 * C2, *tdv = stats + 11 * C2;

  // 1) FPS (serial chain, point cloud resident in 320KB WGP LDS)
  const size_t fpsLds = (size_t)N_PTS * 4 * sizeof(float)
                      + 32 * sizeof(float) + 32 * sizeof(int) + sizeof(int);
  fps_kernel<<<1, 1024, fpsLds, stream>>>(pos, idx);

  // 2) gather centers + emit pos_s / batch outputs
  gather_kernel<<<(M_CTR + 255) / 256, 256, 0, stream>>>(pos, idx, pos_s, out);

  // 3) 32-NN per center (scale 0 uses the first 16)
  const size_t knnLds = (size_t)N_PTS * sizeof(float) + 8 * sizeof(float) + 8 * sizeof(int);
  knn_kernel<<<M_CTR, 256, knnLds, stream>>>(pos, pos_s, nb32);

  // 4) per-scale: features -> WMMA resblock (BN stats fused) -> max-pool
  const int gemmBlocks = (NEDGE / 16) / 8;   // 512 blocks x 8 waves = 4096 row tiles
  for (int i = 0; i < 2; ++i) {
    const int dil = (i == 0) ? 1 : 2;
    feat_kernel<<<NEDGE, 128, 0, stream>>>(x, pos, pos_s, nb32, F, dil);
    hipMemsetAsync(stats, 0, (size_t)6 * C2 * sizeof(float), stream);

    // zero-padded weight copies (branch-free WMMA loops)
    pad_b_kernel<<<(C0P * C1 + 255) / 256, 256, 0, stream>>>(
        w1 + (long)i * C0 * C1, Bp1, C0, C1);
    pad_b_kernel<<<(C0P * C2 + 255) / 256, 256, 0, stream>>>(
        wd + (long)i * C0 * C2, Bpd, C0, C2);

    // H1 = F @ w1 + b1   (67->64), raw + column stats
    wmma_gemm_kernel<C0P / 4, 4, false><<<gemmBlocks, 256, 0, stream>>>(
        F, C0P, Bp1, b1 + (long)i * C1, nullptr, nullptr, H1, sum1, sq1);
    stats_kernel<<<1, 128, 0, stream>>>(sum1, sq1, g1 + (long)i * C1, bt1 + (long)i * C1,
                                        s1v, t1v, C1);

    // H2 = relu(bn1(H1)) @ w2 + b2   (64->128), BN+ReLU fused into A load
    wmma_gemm_kernel<C1 / 4, 8, true><<<gemmBlocks, 256, 0, stream>>>(
        H1, C1, w2 + (long)i * C1 * C2, b2 + (long)i * C2, s1v, t1v, H2, sum2, sq2);

    // Hd = F @ wd + bd   (67->128) shortcut branch
    wmma_gemm_kernel<C0P / 4, 8, false><<<gemmBlocks, 256, 0, stream>>>(
        F, C0P, Bpd, bd + (long)i * C2, nullptr, nullptr, Hd, sumd, sqd);

    stats_kernel<<<1, 128, 0, stream>>>(sum2, sq2, g2 + (long)i * C2, bt2 + (long)i * C2,
                                        s2v, t2v, C2);
    stats_kernel<<<1, 128, 0, stream>>>(sumd, sqd, gd + (long)i * C2, btd + (long)i * C2,
                                        sdv, tdv, C2);

    // relu(bn2(H2)+bnd(Hd)) -> max over K=16 -> x_new[:, i*128:(i+1)*128]
    combine_kernel<<<M_CTR, 128, 0, stream>>>(H2, Hd, s2v, t2v, sdv, tdv, out, i * C2);
  }
}